// CGNN_net_34754875359751
// MI455X (gfx1250) — compile-verified
//
#include <hip/hip_runtime.h>
#include <math.h>

// CGNN pipeline for MI455X (gfx1250, wave32).
// f32 WMMA (v_wmma_f32_16x16x4_f32) for all dense matmuls + cross-attn Gram matrices.

typedef float v2f __attribute__((ext_vector_type(2)));
typedef float v8f __attribute__((ext_vector_type(8)));

#define KNBR 16
#define DOUTF 64

// fast sigmoid: v_exp_f32 + v_rcp_f32 (avoid IEEE div_fixup sequence)
__device__ __forceinline__ float sigf(float x) {
    return __builtin_amdgcn_rcpf(1.0f + __expf(-x));
}
__device__ __forceinline__ float frcp(float x) { return __builtin_amdgcn_rcpf(x); }

// One K-step of the 16-row x 64-col tile: A fragment (2 floats/lane) times
// 4 B tiles taken from W rows k,k+1 at cols {0,16,32,48}+r.
__device__ __forceinline__ void wmma_step(float ax, float ay, const float* __restrict__ wp,
                                          int r, v8f& acc0, v8f& acc1, v8f& acc2, v8f& acc3)
{
    v2f a; a.x = ax; a.y = ay;
    v2f b0; b0.x = wp[ 0 + r]; b0.y = wp[64 +  0 + r];
    v2f b1; b1.x = wp[16 + r]; b1.y = wp[64 + 16 + r];
    v2f b2; b2.x = wp[32 + r]; b2.y = wp[64 + 32 + r];
    v2f b3; b3.x = wp[48 + r]; b3.y = wp[64 + 48 + r];
    acc0 = __builtin_amdgcn_wmma_f32_16x16x4_f32(false, a, false, b0, (short)0, acc0, false, false);
    acc1 = __builtin_amdgcn_wmma_f32_16x16x4_f32(false, a, false, b1, (short)0, acc1, false, false);
    acc2 = __builtin_amdgcn_wmma_f32_16x16x4_f32(false, a, false, b2, (short)0, acc2, false, false);
    acc3 = __builtin_amdgcn_wmma_f32_16x16x4_f32(false, a, false, b3, (short)0, acc3, false, false);
}

// -----------------------------------------------------------------------------
// GEMM: C[N,64] = A[N,128] @ W[128,64] using V_WMMA_F32_16X16X4_F32.
// One wave computes a 16-row block (4 accumulator tiles across 64 cols,
// 32 K-steps of 4). A layout: lane r=lane&15 holds row r; half hi=lane>>4 holds
// K-pair {k+2hi, k+2hi+1}. B mirrored (lane=col, half=K-pair).
// paired: row = sigmoid(cat(A0[N,64], A1[N,64])) built on the fly (h for layer 2),
//         statically split into two half-K loops (no per-iteration select).
// sig_out: apply sigmoid to result (self branch written straight to d_out, ldc=128)
// -----------------------------------------------------------------------------
__global__ __launch_bounds__(256)
void gemm128x64_wmma(const float* __restrict__ A0, const float* __restrict__ A1,
                     const float* __restrict__ W,  float* __restrict__ C,
                     int ldc, int paired, int sig_out, int nrows)
{
    int wave = (blockIdx.x * blockDim.x + threadIdx.x) >> 5;
    int row0 = wave * 16;
    if (row0 >= nrows) return;                  // wave-uniform: EXEC stays all-ones
    int lane = threadIdx.x & 31;
    int r  = lane & 15;
    int hi = lane >> 4;

    v8f acc0 = {0.f,0.f,0.f,0.f,0.f,0.f,0.f,0.f};
    v8f acc1 = acc0, acc2 = acc0, acc3 = acc0;

    if (!paired) {
        const float* ap = A0 + (size_t)(row0 + r) * 128 + 2 * hi;
        for (int k0 = 0; k0 < 128; k0 += 4) {
            const float* p = ap + k0;
            wmma_step(p[0], p[1], W + (size_t)(k0 + 2 * hi) * 64, r, acc0, acc1, acc2, acc3);
        }
    } else {
        // first 64 K from sigmoid(A0), last 64 K from sigmoid(A1); W rows offset by 64
        const float* ap0 = A0 + (size_t)(row0 + r) * 64 + 2 * hi;
        for (int k0 = 0; k0 < 64; k0 += 4) {
            const float* p = ap0 + k0;
            wmma_step(sigf(p[0]), sigf(p[1]), W + (size_t)(k0 + 2 * hi) * 64, r, acc0, acc1, acc2, acc3);
        }
        const float* ap1 = A1 + (size_t)(row0 + r) * 64 + 2 * hi;
        for (int k0 = 0; k0 < 64; k0 += 4) {
            const float* p = ap1 + k0;
            wmma_step(sigf(p[0]), sigf(p[1]), W + (size_t)(64 + k0 + 2 * hi) * 64, r, acc0, acc1, acc2, acc3);
        }
    }

    // D layout: VGPR vg -> M = vg + 8*hi, N = r
    #pragma unroll
    for (int vg = 0; vg < 8; ++vg) {
        int row = row0 + vg + 8 * hi;
        float v0 = acc0[vg], v1 = acc1[vg], v2 = acc2[vg], v3 = acc3[vg];
        if (sig_out) { v0 = sigf(v0); v1 = sigf(v1); v2 = sigf(v2); v3 = sigf(v3); }
        float* cp = C + (size_t)row * ldc;
        cp[ 0 + r] = v0; cp[16 + r] = v1; cp[32 + r] = v2; cp[48 + r] = v3;
    }
}

// -----------------------------------------------------------------------------
// Cross-attention: one wave per node. Gather a1=aggr1[nbr1[i]], a2=aggr2[nbr2[i]]
// (each 16x64). G = a1 @ a2^T via 16 chained f32 WMMAs; lane loads map directly
// onto A(16x4)/B(4x16) layouts. Then d2 = n1[k]+n2[l]-2G, sim=exp(-sqrt(d2)),
// marginals via wave32 shfl reductions.
// -----------------------------------------------------------------------------
__global__ __launch_bounds__(256)
void cross_attn_wmma(const float* __restrict__ aggr1, const int* __restrict__ nbr1,
                     const float* __restrict__ aggr2, const int* __restrict__ nbr2,
                     float* __restrict__ att1, float* __restrict__ att2, int n)
{
    int node = (blockIdx.x * blockDim.x + threadIdx.x) >> 5;
    if (node >= n) return;                      // wave-uniform
    int lane = threadIdx.x & 31;
    int r  = lane & 15;
    int hi = lane >> 4;

    const float* p1 = aggr1 + (size_t)nbr1[node * KNBR + r] * DOUTF + 2 * hi;
    const float* p2 = aggr2 + (size_t)nbr2[node * KNBR + r] * DOUTF + 2 * hi;

    v8f g = {0.f,0.f,0.f,0.f,0.f,0.f,0.f,0.f};
    float sq1 = 0.f, sq2 = 0.f;

    #pragma unroll
    for (int k0 = 0; k0 < DOUTF; k0 += 4) {
        float ax = p1[k0], ay = p1[k0 + 1];
        float bx = p2[k0], by = p2[k0 + 1];
        sq1 += ax * ax + ay * ay;
        sq2 += bx * bx + by * by;
        v2f a; a.x = ax; a.y = ay;
        v2f b; b.x = bx; b.y = by;
        g = __builtin_amdgcn_wmma_f32_16x16x4_f32(false, a, false, b, (short)0, g, false, false);
    }

    // full squared norms per row (halves hold disjoint K ranges)
    float n1 = sq1 + __shfl_xor(sq1, 16, 32);   // lane holds ||a1_{r}||^2
    float n2 = sq2 + __shfl_xor(sq2, 16, 32);   // lane holds ||a2_{r}||^2

    float s[8];
    float local = 0.f;
    #pragma unroll
    for (int vg = 0; vg < 8; ++vg) {
        float n1k = __shfl(n1, vg + 8 * hi, 32);        // norm of row k = vg+8*hi
        float d2  = n1k + n2 - 2.0f * g[vg];
        d2 = fmaxf(d2, 1e-12f);
        float sim = __expf(-sqrtf(d2));
        s[vg] = sim;
        local += sim;
    }

    float total = local;
    #pragma unroll
    for (int m = 1; m < 32; m <<= 1) total += __shfl_xor(total, m, 32);
    float inv = frcp(total);

    // column marginal: sum over all 16 rows of column l=r
    float cs = local + __shfl_xor(local, 16, 32);
    if (hi == 0) att2[(size_t)node * KNBR + r] = cs * inv;

    // row marginals: reduce across the 16 lanes of each half
    #pragma unroll
    for (int vg = 0; vg < 8; ++vg) {
        float rs = s[vg];
        rs += __shfl_xor(rs, 1, 32);
        rs += __shfl_xor(rs, 2, 32);
        rs += __shfl_xor(rs, 4, 32);
        rs += __shfl_xor(rs, 8, 32);
        if (r == vg) att1[(size_t)node * KNBR + vg + 8 * hi] = rs * inv;
    }
}

// -----------------------------------------------------------------------------
// Mean aggregation: aggr[i,d] = (1/K) sum_k xn[nbr[i,k], d]. 64 threads/node,
// coalesced 64-float row gathers (L2 resident: N*64 f32 = 12.8MB << 192MB L2).
// -----------------------------------------------------------------------------
__global__ __launch_bounds__(256)
void aggregate_mean(const float* __restrict__ xn, const int* __restrict__ nbr,
                    float* __restrict__ aggr, int n)
{
    int t = blockIdx.x * blockDim.x + threadIdx.x;
    int node = t >> 6, d = t & 63;
    if (node >= n) return;
    const int* nb = nbr + (size_t)node * KNBR;
    float acc = 0.f;
    #pragma unroll
    for (int k = 0; k < KNBR; ++k)
        acc += xn[(size_t)nb[k] * DOUTF + d];
    aggr[(size_t)node * DOUTF + d] = acc * (1.0f / (float)KNBR);
}

// Weighted aggregation + sigmoid, writes o[:, 64:128] directly into d_out.
__global__ __launch_bounds__(256)
void aggregate_weighted_sig(const float* __restrict__ yn, const int* __restrict__ nbr,
                            const float* __restrict__ ca, const float* __restrict__ sa,
                            float* __restrict__ out, int n)
{
    int t = blockIdx.x * blockDim.x + threadIdx.x;
    int node = t >> 6, d = t & 63;
    if (node >= n) return;
    const int*   nb  = nbr + (size_t)node * KNBR;
    const float* cap = ca  + (size_t)node * KNBR;
    const float* sap = sa  + (size_t)node * KNBR;
    float acc = 0.f;
    #pragma unroll
    for (int k = 0; k < KNBR; ++k)
        acc += cap[k] * sap[k] * yn[(size_t)nb[k] * DOUTF + d];
    out[(size_t)node * 128 + 64 + d] = sigf(acc);
}

// si = aggr @ w[:64], sj = aggr @ w[64:], one thread per node.
__global__ __launch_bounds__(256)
void sisj_kernel(const float* __restrict__ aggr, const float* __restrict__ w_att,
                 float* __restrict__ si, float* __restrict__ sj, int n)
{
    int i = blockIdx.x * blockDim.x + threadIdx.x;
    if (i >= n) return;
    const float* a = aggr + (size_t)i * DOUTF;
    float s0 = 0.f, s1 = 0.f;
    #pragma unroll
    for (int d = 0; d < DOUTF; ++d) {
        float v = a[d];
        s0 += v * w_att[d];
        s1 += v * w_att[DOUTF + d];
    }
    si[i] = s0;
    sj[i] = s1;
}

// sa[i,k] = softmax_k( exp(leaky_relu(si[i] + sj[nbr[i,k]], 0.01)) )
__global__ __launch_bounds__(256)
void self_attn_kernel(const float* __restrict__ si, const float* __restrict__ sj,
                      const int* __restrict__ nbr, float* __restrict__ sa, int n)
{
    int i = blockIdx.x * blockDim.x + threadIdx.x;
    if (i >= n) return;
    float m = si[i];
    const int* nb = nbr + (size_t)i * KNBR;
    float e[KNBR];
    float sum = 0.f;
    #pragma unroll
    for (int k = 0; k < KNBR; ++k) {
        float x = m + sj[nb[k]];
        x = (x > 0.f) ? x : 0.01f * x;
        e[k] = __expf(x);
        sum += e[k];
    }
    float inv = frcp(sum);
    #pragma unroll
    for (int k = 0; k < KNBR; ++k)
        sa[(size_t)i * KNBR + k] = e[k] * inv;
}

// -----------------------------------------------------------------------------
extern "C" void kernel_launch(void* const* d_in, const int* in_sizes, int n_in,
                              void* d_out, int out_size, void* d_ws, size_t ws_size,
                              hipStream_t stream)
{
    const float* x1   = (const float*)d_in[0];
    const float* x2   = (const float*)d_in[1];
    const int*   nbr1 = (const int*)  d_in[2];
    const int*   nbr2 = (const int*)  d_in[3];
    const float* W_e1 = (const float*)d_in[4];
    const float* W_s1 = (const float*)d_in[5];
    const float* W_e2 = (const float*)d_in[6];
    const float* W_s2 = (const float*)d_in[7];
    const float* watt = (const float*)d_in[8];
    float* out = (float*)d_out;

    const int N = in_sizes[0] / 128;            // 50000
    const size_t NF = (size_t)N * DOUTF;

    // workspace layout (floats): ~N*452 floats ~= 90 MB
    float* ws    = (float*)d_ws;
    float* xn1   = ws + 0 * NF;                 // reused as yn1 in layer 2
    float* xn2   = ws + 1 * NF;                 // reused as yn2 in layer 2
    float* xs1   = ws + 2 * NF;
    float* xs2   = ws + 3 * NF;
    float* aggr1 = ws + 4 * NF;
    float* aggr2 = ws + 5 * NF;
    float* si1   = ws + 6 * NF;
    float* sj1   = si1 + N;
    float* si2   = sj1 + N;
    float* sj2   = si2 + N;
    float* sa1   = sj2 + N;
    float* sa2   = sa1 + (size_t)N * KNBR;
    float* ca1   = sa2 + (size_t)N * KNBR;
    float* ca2   = ca1 + (size_t)N * KNBR;

    const int gemm_blocks = (N / 16 + 7) / 8;           // 8 waves/block, wave = 16 rows
    const int agg_blocks  = ((size_t)N * 64 + 255) / 256;
    const int nd_blocks   = (N + 255) / 256;
    const int xat_blocks  = (N + 7) / 8;                // 1 wave/node

    // ---- layer 1: shared weights across both graphs ----
    gemm128x64_wmma<<<gemm_blocks, 256, 0, stream>>>(x1, nullptr, W_e1, xn1, 64, 0, 0, N);
    gemm128x64_wmma<<<gemm_blocks, 256, 0, stream>>>(x1, nullptr, W_s1, xs1, 64, 0, 0, N);
    gemm128x64_wmma<<<gemm_blocks, 256, 0, stream>>>(x2, nullptr, W_e1, xn2, 64, 0, 0, N);
    gemm128x64_wmma<<<gemm_blocks, 256, 0, stream>>>(x2, nullptr, W_s1, xs2, 64, 0, 0, N);

    aggregate_mean<<<agg_blocks, 256, 0, stream>>>(xn1, nbr1, aggr1, N);
    aggregate_mean<<<agg_blocks, 256, 0, stream>>>(xn2, nbr2, aggr2, N);

    // ---- attention stages ----
    sisj_kernel<<<nd_blocks, 256, 0, stream>>>(aggr1, watt, si1, sj1, N);
    sisj_kernel<<<nd_blocks, 256, 0, stream>>>(aggr2, watt, si2, sj2, N);
    self_attn_kernel<<<nd_blocks, 256, 0, stream>>>(si1, sj1, nbr1, sa1, N);
    self_attn_kernel<<<nd_blocks, 256, 0, stream>>>(si2, sj2, nbr2, sa2, N);
    cross_attn_wmma<<<xat_blocks, 256, 0, stream>>>(aggr1, nbr1, aggr2, nbr2, ca1, ca2, N);

    // ---- layer 2: h = sigmoid(cat(xs,aggr)) built on the fly in the GEMM ----
    float* yn1 = xn1;   // xn no longer needed
    float* yn2 = xn2;
    float* o1  = out;                       // [N,128]
    float* o2  = out + (size_t)N * 128;     // [N,128]

    gemm128x64_wmma<<<gemm_blocks, 256, 0, stream>>>(xs1, aggr1, W_e2, yn1, 64, 1, 0, N);
    gemm128x64_wmma<<<gemm_blocks, 256, 0, stream>>>(xs1, aggr1, W_s2, o1, 128, 1, 1, N);
    gemm128x64_wmma<<<gemm_blocks, 256, 0, stream>>>(xs2, aggr2, W_e2, yn2, 64, 1, 0, N);
    gemm128x64_wmma<<<gemm_blocks, 256, 0, stream>>>(xs2, aggr2, W_s2, o2, 128, 1, 1, N);

    aggregate_weighted_sig<<<agg_blocks, 256, 0, stream>>>(yn1, nbr1, ca1, sa1, o1, N);
    aggregate_weighted_sig<<<agg_blocks, 256, 0, stream>>>(yn2, nbr2, ca2, sa2, o2, N);
}